// _fasterRCNN_38173669326957
// MI455X (gfx1250) — compile-verified
//
#include <hip/hip_runtime.h>
#include <hip/hip_bf16.h>
#include <stdint.h>

#define N_ROIS    3000
#define N_CLASSES 21
#define FEAT_H    64
#define FEAT_W    128
#define PROB_STRIDE 24            // padded row stride for aligned float2 stores
#define N_BOX_PAD 3008            // 3000 rounded up to multiple of 32
#define SCALE     0.0625f         // 128/2048 == 64/1024 == 1/16 exactly

typedef float v2f __attribute__((ext_vector_type(2)));
typedef float v8f __attribute__((ext_vector_type(8)));

// ---------------------------------------------------------------------------
// Kernel 1: softmax of cls_score rows (16 ROIs per wave) with row-sums done by
// V_WMMA_F32_16X16X4_F32 (exact f32), plus box quantization into packed u32.
// A-matrix layout (32-bit, 16x4): lanes 0-15 hold {K=0,K=1} of row M=lane,
// lanes 16-31 hold {K=2,K=3} of row M=lane-16.  We assign columns so the exp
// values land directly in that layout: chunk kc covers cols 4kc..4kc+3.
// ---------------------------------------------------------------------------
__global__ __launch_bounds__(256)
void k_softmax_boxes(const float* __restrict__ rois,
                     const float* __restrict__ cls,
                     float* __restrict__ probs,      // [N_ROIS][PROB_STRIDE]
                     unsigned* __restrict__ boxes)   // [N_BOX_PAD]
{
    const int tid   = blockIdx.x * blockDim.x + threadIdx.x;
    const int lane  = threadIdx.x & 31;
    const int gwave = tid >> 5;
    const int rloc  = lane & 15;          // row within the 16-ROI tile
    const int half  = lane >> 4;          // 0: cols 4k,4k+1   1: cols 4k+2,4k+3
    const int r     = gwave * 16 + rloc;  // global ROI index
    const bool rv   = (r < N_ROIS);

    // ---- load my 12 score entries (6 chunks x 2 adjacent columns) ----
    float s[6][2];
    float m_part = -3.4e38f;
#pragma unroll
    for (int kc = 0; kc < 6; ++kc) {
        const int c0 = 4 * kc + 2 * half;
#pragma unroll
        for (int j = 0; j < 2; ++j) {
            const int c = c0 + j;
            float v = -3.4e38f;
            if (rv && c < N_CLASSES) v = cls[r * N_CLASSES + c];
            s[kc][j] = v;
            m_part = fmaxf(m_part, v);
        }
    }
    // full row max: halves of the row live in lanes L and L^16
    const float m = fmaxf(m_part, __shfl_xor(m_part, 16));

    // ---- exp with zero padding for cols >= 21 ----
    float e[6][2];
#pragma unroll
    for (int kc = 0; kc < 6; ++kc) {
        const int c0 = 4 * kc + 2 * half;
#pragma unroll
        for (int j = 0; j < 2; ++j) {
            const int c = c0 + j;
            e[kc][j] = (c < N_CLASSES) ? __expf(s[kc][j] - m) : 0.0f;
        }
    }

    // ---- row sums via 6 chained f32 WMMAs: D = E(16x4) * ones(4x16) + C ----
    v8f acc = {};
    const v2f bones = {1.0f, 1.0f};
#pragma unroll
    for (int kc = 0; kc < 6; ++kc) {
        v2f a = {e[kc][0], e[kc][1]};
        acc = __builtin_amdgcn_wmma_f32_16x16x4_f32(
            false, a, false, bones, (short)0, acc, false, false);
    }
    // D layout: lanes 0-15 VGPR i = sum[row i] (i=0..7); lanes 16-31 = rows 8..15.
    const int rsel = rloc & 7;
    float localv = acc[0];
    localv = (rsel == 1) ? acc[1] : localv;
    localv = (rsel == 2) ? acc[2] : localv;
    localv = (rsel == 3) ? acc[3] : localv;
    localv = (rsel == 4) ? acc[4] : localv;
    localv = (rsel == 5) ? acc[5] : localv;
    localv = (rsel == 6) ? acc[6] : localv;
    localv = (rsel == 7) ? acc[7] : localv;
    const float remotev = __shfl_xor(localv, 16);
    const bool useLocal = (((rloc >> 3) & 1) == half);
    const float sum = useLocal ? localv : remotev;
    const float inv = 1.0f / sum;

    // ---- store probabilities (padded stride, aligned float2 where possible) ----
    if (rv) {
#pragma unroll
        for (int kc = 0; kc < 6; ++kc) {
            const int c0 = 4 * kc + 2 * half;
            if (c0 + 1 < N_CLASSES) {
                float2 p = make_float2(e[kc][0] * inv, e[kc][1] * inv);
                *(float2*)(probs + r * PROB_STRIDE + c0) = p;
            } else if (c0 < N_CLASSES) {
                probs[r * PROB_STRIDE + c0] = e[kc][0] * inv;
            }
        }
    }

    // ---- quantize + pack boxes: one dword per ROI (x:7 bits, y:6 bits) ----
    if (half == 0 && rv) {
        const float* rr = rois + r * 5;
        unsigned x1 = (unsigned)(int)floorf(rr[1] * SCALE);
        unsigned y1 = (unsigned)(int)floorf(rr[2] * SCALE);
        unsigned x2 = (unsigned)(int)floorf(rr[3] * SCALE);
        unsigned y2 = (unsigned)(int)floorf(rr[4] * SCALE);
        boxes[r] = (x1 & 0xFF) | ((y1 & 0xFF) << 8) | ((x2 & 0xFF) << 16) | (y2 << 24);
    }
    // sentinel padding entries: x1=255 never covers any pixel (x <= 127)
    if (tid < (N_BOX_PAD - N_ROIS)) boxes[N_ROIS + tid] = 0x000000FFu;
}

// ---------------------------------------------------------------------------
// Kernel 2: stage packed box table into LDS with CDNA5 async global->LDS
// copies, then per-pixel descending scan (32 boxes/iter via wave32 ballot),
// gather winner's softmax row, write output.
// ---------------------------------------------------------------------------
__global__ __launch_bounds__(256)
void k_scan_gather(const unsigned* __restrict__ boxes,
                   const float* __restrict__ probs,
                   float* __restrict__ out)
{
    __shared__ unsigned sbox[N_BOX_PAD];

    // async copy: 3008 dwords = 752 x b128, 256 threads
    for (int i = threadIdx.x; i < N_BOX_PAD / 4; i += 256) {
        unsigned lds_off = (unsigned)(uintptr_t)(&sbox[i * 4]);
        uint64_t gaddr   = (uint64_t)(uintptr_t)(boxes + i * 4);
        asm volatile("global_load_async_to_lds_b128 %0, %1, off"
                     :: "v"(lds_off), "v"(gaddr) : "memory");
    }
    asm volatile("s_wait_asynccnt 0x0" ::: "memory");
    __syncthreads();

    const int lane  = threadIdx.x & 31;
    const int gwave = (blockIdx.x * blockDim.x + threadIdx.x) >> 5; // 0..511

#pragma unroll 1
    for (int i = 0; i < 16; ++i) {
        const int pixel = gwave * 16 + i;          // 0..8191
        const int px = pixel & (FEAT_W - 1);
        const int py = pixel >> 7;

        int winner = -1;
#pragma unroll 1
        for (int c = (N_BOX_PAD / 32) - 1; c >= 0; --c) {
            const unsigned b = sbox[c * 32 + lane];
            const int x1 = (int)(b & 0xFFu);
            const int y1 = (int)((b >> 8) & 0xFFu);
            const int x2 = (int)((b >> 16) & 0xFFu);
            const int y2 = (int)(b >> 24);
            const bool cov = (px >= x1) & (px <= x2) & (py >= y1) & (py <= y2);
            const unsigned mask = (unsigned)__ballot(cov);
            if (mask) {                            // uniform across the wave
                winner = c * 32 + (31 - __builtin_clz(mask));
                break;
            }
        }

        float val = 0.0f;
        if (winner >= 0 && lane < N_CLASSES)
            val = probs[winner * PROB_STRIDE + lane];
        if (lane < N_CLASSES)
            out[pixel * N_CLASSES + lane] = val;
    }
}

// ---------------------------------------------------------------------------
extern "C" void kernel_launch(void* const* d_in, const int* in_sizes, int n_in,
                              void* d_out, int out_size, void* d_ws, size_t ws_size,
                              hipStream_t stream)
{
    const float* rois = (const float*)d_in[0];   // [3000,5]
    const float* cls  = (const float*)d_in[1];   // [3000,21]
    float* out        = (float*)d_out;           // [8192,21]

    float*    probs = (float*)d_ws;                                       // 288000 B
    unsigned* boxes = (unsigned*)((char*)d_ws + N_ROIS * PROB_STRIDE * 4); // 12032 B

    // 3072 ROIs of capacity -> 24 blocks x 8 waves x 16 ROIs
    k_softmax_boxes<<<24, 256, 0, stream>>>(rois, cls, probs, boxes);
    // 8192 pixels -> 512 waves x 16 pixels -> 64 blocks
    k_scan_gather<<<64, 256, 0, stream>>>(boxes, probs, out);
}